// AdvClassDiscriminator_9148280341127
// MI455X (gfx1250) — compile-verified
//
#include <hip/hip_runtime.h>

typedef float v2f __attribute__((ext_vector_type(2)));
typedef float v8f __attribute__((ext_vector_type(8)));

#define BN_EPS 1e-5f

// ---------------------------------------------------------------------------
// Farthest point sampling: one block per batch, iterative argmax of min-dist.
// mind is a global scratch row [N] per batch. Matches jax scan semantics:
// outputs the *previous* "far" each step, starting at index 0.
// Tie-break: lowest index (jnp.argmax picks first maximum).
// ---------------------------------------------------------------------------
__global__ void fps_kernel(const float* __restrict__ xyz, float* __restrict__ mind,
                           int* __restrict__ outIdx, float* __restrict__ newxyz,
                           int N, int npoint)
{
    int b = blockIdx.x, t = threadIdx.x;
    const float* p = xyz + (size_t)b * N * 3;
    float* md = mind + (size_t)b * N;
    __shared__ float sv[256];
    __shared__ int   si[256];
    __shared__ int   sfar;
    __shared__ float sc[3];
    for (int i = t; i < N; i += 256) md[i] = 1e10f;
    if (t == 0) sfar = 0;
    __syncthreads();
    for (int j = 0; j < npoint; ++j) {
        int far = sfar;
        if (t == 0) {
            outIdx[b * npoint + j] = far;
            float cx = p[far * 3 + 0], cy = p[far * 3 + 1], cz = p[far * 3 + 2];
            sc[0] = cx; sc[1] = cy; sc[2] = cz;
            float* nx = newxyz + ((size_t)b * npoint + j) * 3;
            nx[0] = cx; nx[1] = cy; nx[2] = cz;
        }
        __syncthreads();
        float cx = sc[0], cy = sc[1], cz = sc[2];
        float best = -1.0f; int bi = 0;
        for (int i = t; i < N; i += 256) {
            float dx = p[i * 3 + 0] - cx;
            float dy = p[i * 3 + 1] - cy;
            float dz = p[i * 3 + 2] - cz;
            float d = dx * dx + dy * dy + dz * dz;
            float m = md[i];
            m = fminf(m, d);
            md[i] = m;
            if (m > best) { best = m; bi = i; }   // strict > keeps lowest index
        }
        sv[t] = best; si[t] = bi;
        __syncthreads();
        for (int s = 128; s > 0; s >>= 1) {
            if (t < s) {
                if (sv[t + s] > sv[t] || (sv[t + s] == sv[t] && si[t + s] < si[t])) {
                    sv[t] = sv[t + s]; si[t] = si[t + s];
                }
            }
            __syncthreads();
        }
        if (t == 0) sfar = si[0];
        __syncthreads();
    }
}

// ---------------------------------------------------------------------------
// Ball query: one thread per center; first K indices (ascending) with d2<r2,
// pad with first hit; if no hit -> N-1 (matches sort/where/min semantics).
// ---------------------------------------------------------------------------
__global__ void ballquery_kernel(const float* __restrict__ xyz, const float* __restrict__ newxyz,
                                 int* __restrict__ idx, int B, int N, int M, float r2, int K)
{
    int g = blockIdx.x * blockDim.x + threadIdx.x;
    if (g >= B * M) return;
    int b = g / M, m = g - b * M;
    const float* p = xyz + (size_t)b * N * 3;
    const float* c = newxyz + ((size_t)b * M + m) * 3;
    float cx = c[0], cy = c[1], cz = c[2];
    int* o = idx + ((size_t)b * M + m) * K;
    int cnt = 0;
    for (int i = 0; i < N && cnt < K; ++i) {
        float dx = p[i * 3 + 0] - cx;
        float dy = p[i * 3 + 1] - cy;
        float dz = p[i * 3 + 2] - cz;
        float d = dx * dx + dy * dy + dz * dz;
        if (d < r2) { o[cnt++] = i; }
    }
    if (cnt == 0) {
        for (int k = 0; k < K; ++k) o[k] = N - 1;
    } else {
        int f0 = o[0];
        for (int k = cnt; k < K; ++k) o[k] = f0;
    }
}

// ---------------------------------------------------------------------------
// Grouping: X[CtotPad x (B*M*K)] channel-major, rows >= C+3 zero-filled.
// c<3  : (xyz[idx]-center)*invR
// c>=3 : feats at index  (c-3)*featCS + b*featBS + idx
// ---------------------------------------------------------------------------
__global__ void group_kernel(const float* __restrict__ xyz, const float* __restrict__ newxyz,
                             const int* __restrict__ idx, const float* __restrict__ feats,
                             int featCS, int featBS, float* __restrict__ X,
                             int B, int N, int M, int K, int C, int CtotPad, float invR)
{
    int Nt = B * M * K;
    long long total = (long long)CtotPad * Nt;
    long long stride = (long long)gridDim.x * blockDim.x;
    for (long long g = blockIdx.x * (long long)blockDim.x + threadIdx.x; g < total; g += stride) {
        int c = (int)(g / Nt);
        int n = (int)(g - (long long)c * Nt);
        float v = 0.0f;
        if (c < C + 3) {
            int k = n % K;
            int bm = n / K;
            int m = bm % M;
            int b = bm / M;
            int pi = idx[((size_t)b * M + m) * K + k];
            if (c < 3) {
                v = (xyz[((size_t)b * N + pi) * 3 + c] - newxyz[((size_t)b * M + m) * 3 + c]) * invR;
            } else {
                v = feats[(size_t)(c - 3) * featCS + (size_t)b * featBS + pi];
            }
        }
        X[(size_t)c * Nt + n] = v;
    }
}

// ---------------------------------------------------------------------------
// Zero-pad W [Cout x Cin] -> Wp [CoutPad x CinPad], bias -> bp [CoutPad].
// ---------------------------------------------------------------------------
__global__ void pad_params_kernel(const float* __restrict__ Wsrc, const float* __restrict__ bsrc,
                                  float* __restrict__ Wdst, float* __restrict__ bdst,
                                  int Cout, int Cin, int CoutPad, int CinPad)
{
    int total = CoutPad * CinPad + CoutPad;
    int stride = gridDim.x * blockDim.x;
    for (int g = blockIdx.x * blockDim.x + threadIdx.x; g < total; g += stride) {
        if (g < CoutPad * CinPad) {
            int r = g / CinPad, c = g - r * CinPad;
            Wdst[g] = (r < Cout && c < Cin) ? Wsrc[r * Cin + c] : 0.0f;
        } else {
            int r = g - CoutPad * CinPad;
            bdst[r] = (r < Cout) ? bsrc[r] : 0.0f;
        }
    }
}

// ---------------------------------------------------------------------------
// GEMM + bias via V_WMMA_F32_16X16X4_F32. One wave per 16x32 output region.
// Y[CoutPad x NT] = Wp[CoutPad x CINP] * X[CINP x NT] + bp.
// All dims padded -> no conditional loads; K-loop fully unrolled
// (compile-time CINP/NT => constant address offsets).
// A frag: (m=lane&15, k=2*(lane>>4)+v)   B frag: (k=2*(lane>>4)+v, n=lane&15)
// D frag: (m=v+8*(lane>>4), n=lane&15)
// ---------------------------------------------------------------------------
template <int CINP, int NT>
__global__ void gemm_bias_wmma(const float* __restrict__ Wp, const float* __restrict__ bp,
                               const float* __restrict__ X, float* __restrict__ Y)
{
    const int nBase = blockIdx.x * 32;
    const int oBase = blockIdx.y * 16;
    const int lane  = threadIdx.x;
    const int half  = lane >> 4;
    const int l15   = lane & 15;

    const float* wrow = Wp + (oBase + l15) * CINP + 2 * half;   // 8B aligned (CINP%4==0)
    const float* xh   = X + nBase + l15 + 2 * half * NT;

    v8f acc0 = {};
    v8f acc1 = {};
#pragma unroll
    for (int kc = 0; kc < CINP; kc += 4) {
        v2f a = *(const v2f*)(wrow + kc);
        v2f b0, b1;
        b0.x = xh[kc * NT];
        b0.y = xh[kc * NT + NT];
        b1.x = xh[kc * NT + 16];
        b1.y = xh[kc * NT + NT + 16];
        acc0 = __builtin_amdgcn_wmma_f32_16x16x4_f32(false, a, false, b0, (short)0, acc0, false, false);
        acc1 = __builtin_amdgcn_wmma_f32_16x16x4_f32(false, a, false, b1, (short)0, acc1, false, false);
    }
#pragma unroll
    for (int r = 0; r < 8; ++r) {
        int o = oBase + r + 8 * half;
        float bv = bp[o];
        Y[o * NT + nBase + l15]      = acc0[r] + bv;
        Y[o * NT + nBase + l15 + 16] = acc1[r] + bv;
    }
}

// ---------------------------------------------------------------------------
// Fused per-channel BatchNorm (train mode: batch statistics) + ReLU, in place.
// One block per (real) channel; two passes over the row.
// ---------------------------------------------------------------------------
__global__ void bn_relu_kernel(float* __restrict__ Y, const float* __restrict__ g,
                               const float* __restrict__ bt, int Nt)
{
    int c = blockIdx.x, t = threadIdx.x;
    float* row = Y + (size_t)c * Nt;
    __shared__ float ss[256], sq[256];
    __shared__ float mi[2];
    float s = 0.0f, q = 0.0f;
    for (int i = t; i < Nt; i += 256) { float v = row[i]; s += v; q += v * v; }
    ss[t] = s; sq[t] = q;
    __syncthreads();
    for (int st = 128; st > 0; st >>= 1) {
        if (t < st) { ss[t] += ss[t + st]; sq[t] += sq[t + st]; }
        __syncthreads();
    }
    if (t == 0) {
        float mean = ss[0] / (float)Nt;
        float var = sq[0] / (float)Nt - mean * mean;
        mi[0] = mean;
        mi[1] = rsqrtf(var + BN_EPS);
    }
    __syncthreads();
    float mean = mi[0], inv = mi[1], gg = g[c], bb = bt[c];
    for (int i = t; i < Nt; i += 256) {
        float v = (row[i] - mean) * inv * gg + bb;
        row[i] = v > 0.0f ? v : 0.0f;
    }
}

// ---------------------------------------------------------------------------
// Max pool over K: F[c][j] = max_k Y[c][j*K + k],  j = b*M+m in [0, BM)
// ---------------------------------------------------------------------------
__global__ void maxpool_kernel(const float* __restrict__ Y, float* __restrict__ F,
                               int C, int BM, int K)
{
    int g = blockIdx.x * blockDim.x + threadIdx.x;
    if (g >= C * BM) return;
    int c = g / BM, j = g - c * BM;
    const float* r = Y + (size_t)c * BM * K + (size_t)j * K;
    float mx = r[0];
    for (int k = 1; k < K; ++k) mx = fmaxf(mx, r[k]);
    F[(size_t)c * BM + j] = mx;
}

// ---------------------------------------------------------------------------
// Head: out[b] = sum_m ( mean_c F[c][b*64+m] ) * w[m] + fcb
// F is [64][B*64]. One block per batch, 64 threads (one per m).
// ---------------------------------------------------------------------------
__global__ void head_kernel(const float* __restrict__ F, const float* __restrict__ w,
                            const float* __restrict__ fcb, float* __restrict__ out, int B)
{
    int b = blockIdx.x, m = threadIdx.x; // 64 threads
    float s = 0.0f;
    for (int c = 0; c < 64; ++c) s += F[(size_t)c * (B * 64) + b * 64 + m];
    s = (s * (1.0f / 64.0f)) * w[m];
    __shared__ float sm[64];
    sm[m] = s;
    __syncthreads();
    for (int st = 32; st > 0; st >>= 1) {
        if (m < st) sm[m] += sm[m + st];
        __syncthreads();
    }
    if (m == 0) out[b] = sm[0] + fcb[0];
}

// ---------------------------------------------------------------------------
// Host helpers
// ---------------------------------------------------------------------------
static inline int pad16(int x) { return (x + 15) & ~15; }
static inline int pad4(int x)  { return (x + 3) & ~3; }

static void launch_gemm(const float* Wp, const float* bp, const float* X, float* Y,
                        int CoutPad, int CinPad, int Nt, hipStream_t st)
{
    dim3 grid(Nt / 32, CoutPad / 16);
    if (Nt == 32768) {
        if (CinPad == 68)      gemm_bias_wmma<68, 32768><<<grid, 32, 0, st>>>(Wp, bp, X, Y);
        else                   gemm_bias_wmma<32, 32768><<<grid, 32, 0, st>>>(Wp, bp, X, Y);
    } else {
        if (CinPad == 72)      gemm_bias_wmma<72, 16384><<<grid, 32, 0, st>>>(Wp, bp, X, Y);
        else                   gemm_bias_wmma<32, 16384><<<grid, 32, 0, st>>>(Wp, bp, X, Y);
    }
}

// Run one 3-sublayer shared MLP (GEMM -> BN -> ReLU), ping-pong X -> A -> B -> X.
static void mlp3(float* const Wp[3], float* const bp[3],
                 const float* const gs[3], const float* const ts[3],
                 const int w[4], float* X, float* A, float* Bf, int Nt, hipStream_t st)
{
    float* ins[3]  = { X, A, Bf };
    float* outs[3] = { A, Bf, X };
    for (int l = 0; l < 3; ++l) {
        int co = w[l + 1];
        launch_gemm(Wp[l], bp[l], ins[l], outs[l], pad16(co), pad4(w[l]), Nt, st);
        bn_relu_kernel<<<co, 256, 0, st>>>(outs[l], gs[l], ts[l], Nt);
    }
}

extern "C" void kernel_launch(void* const* d_in, const int* in_sizes, int n_in,
                              void* d_out, int out_size, void* d_ws, size_t ws_size,
                              hipStream_t stream)
{
    (void)in_sizes; (void)n_in; (void)out_size; (void)ws_size;

    const float* xyz   = (const float*)d_in[0];
    const float* feats = (const float*)d_in[1];
    // params1 at 2..13, params2 at 14..25, params3 at 26..37 (W,b,g,bt per sub)
    const float* P[3][4][3]; // [layer][quad][sub]
    for (int L = 0; L < 3; ++L)
        for (int s = 0; s < 3; ++s)
            for (int q = 0; q < 4; ++q)
                P[L][q][s] = (const float*)d_in[2 + L * 12 + s * 4 + q];
    const float* fcw = (const float*)d_in[38];
    const float* fcb = (const float*)d_in[39];
    float* out = (float*)d_out;

    const int B = 16, N1 = 16384, M1 = 128, M2 = 64, M3 = 64, K = 16;

    // Per-sublayer (Cout, Cin)
    static const int dims[3][3][2] = {
        { {32, 67}, {32, 32}, {67, 32} },
        { {32, 70}, {32, 32}, {67, 32} },
        { {32, 70}, {32, 32}, {64, 32} },
    };

    // workspace bump allocator (256B aligned)
    char* base = (char*)d_ws;
    size_t off = 0;
    auto alloc = [&](size_t bytes) -> void* {
        void* p = base + off;
        off = (off + bytes + 255) & ~(size_t)255;
        return p;
    };
    float* mind = (float*)alloc((size_t)B * N1 * 4);
    int*   fpsi = (int*)  alloc((size_t)B * M1 * 4);
    float* nx1  = (float*)alloc((size_t)B * M1 * 3 * 4);
    int*   id1  = (int*)  alloc((size_t)B * M1 * K * 4);
    float* nx2  = (float*)alloc((size_t)B * M2 * 3 * 4);
    int*   id2  = (int*)  alloc((size_t)B * M2 * K * 4);
    float* nx3  = (float*)alloc((size_t)B * M3 * 3 * 4);
    int*   id3  = (int*)  alloc((size_t)B * M3 * K * 4);
    float* Xb   = (float*)alloc((size_t)80 * 32768 * 4);   // holds up to CoutPad=80 rows
    float* Ab   = (float*)alloc((size_t)32 * 32768 * 4);
    float* Bb   = (float*)alloc((size_t)32 * 32768 * 4);
    float* f1   = (float*)alloc((size_t)67 * B * M1 * 4);
    float* f2   = (float*)alloc((size_t)67 * B * M2 * 4);
    float* f3   = (float*)alloc((size_t)64 * B * M3 * 4);

    // Padded params
    float* Wp[3][3];
    float* bp[3][3];
    for (int L = 0; L < 3; ++L) {
        for (int s = 0; s < 3; ++s) {
            int cop = pad16(dims[L][s][0]), cip = pad4(dims[L][s][1]);
            Wp[L][s] = (float*)alloc((size_t)cop * cip * 4);
            bp[L][s] = (float*)alloc((size_t)cop * 4);
            pad_params_kernel<<<8, 256, 0, stream>>>(P[L][0][s], P[L][1][s], Wp[L][s], bp[L][s],
                                                     dims[L][s][0], dims[L][s][1], cop, cip);
        }
    }

    // -------------------- Layer 1 --------------------
    fps_kernel<<<B, 256, 0, stream>>>(xyz, mind, fpsi, nx1, N1, M1);
    ballquery_kernel<<<(B * M1 + 255) / 256, 256, 0, stream>>>(xyz, nx1, id1, B, N1, M1, 1.0f, K);
    int Nt1 = B * M1 * K; // 32768
    group_kernel<<<1024, 256, 0, stream>>>(xyz, nx1, id1, feats,
                                           /*featCS=*/N1, /*featBS=*/64 * N1,
                                           Xb, B, N1, M1, K, /*C=*/64, /*CtotPad=*/68, 1.0f);
    {
        const int w[4] = { 67, 32, 32, 67 };
        mlp3(Wp[0], bp[0], P[0][2], P[0][3], w, Xb, Ab, Bb, Nt1, stream);
    }
    maxpool_kernel<<<(67 * B * M1 + 255) / 256, 256, 0, stream>>>(Xb, f1, 67, B * M1, K);

    // -------------------- Layer 2 --------------------
    fps_kernel<<<B, 256, 0, stream>>>(nx1, mind, fpsi, nx2, M1, M2);
    ballquery_kernel<<<(B * M2 + 255) / 256, 256, 0, stream>>>(nx1, nx2, id2, B, M1, M2, 1.0f, K);
    int Nt2 = B * M2 * K; // 16384
    group_kernel<<<512, 256, 0, stream>>>(nx1, nx2, id2, f1,
                                          /*featCS=*/B * M1, /*featBS=*/M1,
                                          Xb, B, M1, M2, K, /*C=*/67, /*CtotPad=*/72, 1.0f);
    {
        const int w[4] = { 70, 32, 32, 67 };
        mlp3(Wp[1], bp[1], P[1][2], P[1][3], w, Xb, Ab, Bb, Nt2, stream);
    }
    maxpool_kernel<<<(67 * B * M2 + 255) / 256, 256, 0, stream>>>(Xb, f2, 67, B * M2, K);

    // -------------------- Layer 3 --------------------
    fps_kernel<<<B, 256, 0, stream>>>(nx2, mind, fpsi, nx3, M2, M3);
    ballquery_kernel<<<(B * M3 + 255) / 256, 256, 0, stream>>>(nx2, nx3, id3, B, M2, M3, 1.0f, K);
    int Nt3 = B * M3 * K; // 16384
    group_kernel<<<512, 256, 0, stream>>>(nx2, nx3, id3, f2,
                                          /*featCS=*/B * M2, /*featBS=*/M2,
                                          Xb, B, M2, M3, K, /*C=*/67, /*CtotPad=*/72, 1.0f);
    {
        const int w[4] = { 70, 32, 32, 64 };
        mlp3(Wp[2], bp[2], P[2][2], P[2][3], w, Xb, Ab, Bb, Nt3, stream);
    }
    maxpool_kernel<<<(64 * B * M3 + 255) / 256, 256, 0, stream>>>(Xb, f3, 64, B * M3, K);

    // -------------------- Head --------------------
    head_kernel<<<B, 64, 0, stream>>>(f3, fcw, fcb, out, B);
}